// DeepSeekMoE_11725260718230
// MI455X (gfx1250) — compile-verified
//
#include <hip/hip_runtime.h>
#include <hip/hip_bf16.h>
#include <math.h>

typedef __attribute__((ext_vector_type(2))) float v2f;
typedef __attribute__((ext_vector_type(8))) float v8f;

__device__ __forceinline__ v8f wmma4(v2f a, v2f b, v8f c) {
  // D = A(16x4) * B(4x16) + C(16x16), fp32 native WMMA
  return __builtin_amdgcn_wmma_f32_16x16x4_f32(
      /*neg_a=*/false, a, /*neg_b=*/false, b,
      /*c_mod=*/(short)0, c, /*reuse_a=*/false, /*reuse_b=*/false);
}

#define TM   32     // tokens per block
#define XSS  260    // x tile LDS stride (pad: 260%64==4 -> conflict-free b64 A loads)
#define HSS  132    // h tile LDS stride

// ---------------------------------------------------------------------------
// Fold conv1d (4 ch, k=3, pad 1) + flatten + C_W into an effective 256->128 GEMM
// ---------------------------------------------------------------------------
__global__ __launch_bounds__(128) void prep_c_kernel(
    const float* __restrict__ C_cw, const float* __restrict__ C_cb,
    const float* __restrict__ C_W,  const float* __restrict__ C_b,
    float* __restrict__ Weff, float* __restrict__ beff)
{
  const int e = blockIdx.x;      // expert 0/1
  const int p = threadIdx.x;     // output row 0..127
  const float* cw = C_cw + e * 12;          // [c][k] = cw[c*3+k]
  const float* cb = C_cb + e * 4;
  const float* W  = C_W + ((size_t)e * 128 + p) * 1024;  // [c*256+pos]
  // effective bias: C_b + sum_c cb[c]*sum_pos W[p][c*256+pos]
  float bsum = C_b[e * 128 + p];
  for (int c = 0; c < 4; ++c) {
    float rs = 0.f;
    for (int pos = 0; pos < 256; ++pos) rs += W[c * 256 + pos];
    bsum += cb[c] * rs;
  }
  beff[e * 128 + p] = bsum;
  float* wo = Weff + ((size_t)e * 128 + p) * 256;
  for (int i = 0; i < 256; ++i) {
    float s = 0.f;
    for (int c = 0; c < 4; ++c)
      for (int k = 0; k < 3; ++k) {
        int pos = i + 1 - k;                 // conv[c][pos] uses x[pos-1+k]
        if (pos >= 0 && pos < 256) s += cw[c * 3 + k] * W[c * 256 + pos];
      }
    wo[i] = s;
  }
}

// ---------------------------------------------------------------------------
// Gating: gh = tanh([x,mod]@G_W1^T + b1) via fp32 WMMA; logits; top-4 softmax
// ---------------------------------------------------------------------------
__global__ __launch_bounds__(256) void gate_kernel(
    const float* __restrict__ x, const int* __restrict__ modality,
    const float* __restrict__ GW1, const float* __restrict__ Gb1,
    const float* __restrict__ GW2, const float* __restrict__ Gb2,
    float* __restrict__ gatew, float* __restrict__ probs)
{
  __shared__ float xs[TM * XSS];
  __shared__ float mods[TM];
  __shared__ float gh[TM * HSS];
  __shared__ float lg[TM * 8];

  const int tid = threadIdx.x;
  const int b0  = blockIdx.x * TM;

  for (int i = tid; i < TM * 256; i += 256) {
    int m = i >> 8, k = i & 255;
    xs[m * XSS + k] = x[(size_t)(b0 + m) * 256 + k];
  }
  if (tid < TM) mods[tid] = (float)modality[b0 + tid];
  __syncthreads();

  const int lane = tid & 31, wave = tid >> 5;
  const int ln = lane & 15, hi = lane >> 4;
  const int n = wave * 16 + ln;             // gh column 0..127

  v8f c0 = {}; v8f c1 = {};
  const float* wrow = GW1 + (size_t)n * 257;
  for (int k0 = 0; k0 < 256; k0 += 4) {
    const int ko = k0 + 2 * hi;
    v2f a0 = *(const v2f*)&xs[ln * XSS + ko];
    v2f a1 = *(const v2f*)&xs[(16 + ln) * XSS + ko];
    v2f bb; bb.x = wrow[ko]; bb.y = wrow[ko + 1];   // stride 257 (odd) -> scalar loads
    c0 = wmma4(a0, bb, c0);
    c1 = wmma4(a1, bb, c1);
  }
  const float wmod = wrow[256];
  const float bias = Gb1[n];
  for (int r = 0; r < 8; ++r) {
    int m0 = r + 8 * hi;
    gh[m0 * HSS + n]        = tanhf(c0[r] + mods[m0] * wmod + bias);
    gh[(16 + m0) * HSS + n] = tanhf(c1[r] + mods[16 + m0] * wmod + bias);
  }
  __syncthreads();

  { // logits: (32 tok) x (8 experts)
    int tok = tid >> 3, e = tid & 7;
    const float* w2 = GW2 + e * 128;
    float s = Gb2[e];
    for (int o = 0; o < 128; ++o) s += gh[tok * HSS + o] * w2[o];
    s *= (1.0f / 0.7f);
    s = fminf(10.0f, fmaxf(-10.0f, s));
    lg[tok * 8 + e] = s;
  }
  __syncthreads();

  if (tid < TM) {
    float l[8];
    for (int e = 0; e < 8; ++e) l[e] = lg[tid * 8 + e];
    int   idx[4]; float val[4];
    bool  used[8] = {false, false, false, false, false, false, false, false};
    for (int j = 0; j < 4; ++j) {           // top-4, first index wins ties
      int bi = -1; float bv = -1e30f;
      for (int e = 0; e < 8; ++e)
        if (!used[e] && l[e] > bv) { bv = l[e]; bi = e; }
      idx[j] = bi; val[j] = bv; used[bi] = true;
    }
    const float mx = val[0];                // global max too
    float w4[4], s4 = 0.f;
    for (int j = 0; j < 4; ++j) { w4[j] = expf(val[j] - mx); s4 += w4[j]; }
    float gw8[8] = {0.f, 0.f, 0.f, 0.f, 0.f, 0.f, 0.f, 0.f};
    for (int j = 0; j < 4; ++j) gw8[idx[j]] = w4[j] / s4;
    float p8[8], ps = 0.f;
    for (int e = 0; e < 8; ++e) { p8[e] = expf(l[e] - mx); ps += p8[e]; }
    const size_t base = (size_t)(b0 + tid) * 8;
    for (int e = 0; e < 8; ++e) {
      gatew[base + e] = gw8[e];
      probs[base + e] = p8[e] / ps;
    }
  }
}

// ---------------------------------------------------------------------------
// usage[e] = mean_b probs[b][e]; aux KL term
// ---------------------------------------------------------------------------
__global__ __launch_bounds__(256) void usage_kernel(
    const float* __restrict__ probs, float* __restrict__ usage, int Btot)
{
  __shared__ float red[256];
  const int e = blockIdx.x, tid = threadIdx.x;
  float s = 0.f;
  for (int b = tid; b < Btot; b += 256) s += probs[(size_t)b * 8 + e];
  red[tid] = s; __syncthreads();
  for (int off = 128; off > 0; off >>= 1) {
    if (tid < off) red[tid] += red[tid + off];
    __syncthreads();
  }
  if (tid == 0) usage[e] = red[0] / (float)Btot;
}

__global__ void aux_kernel(const float* __restrict__ usage, float* __restrict__ out_aux)
{
  if (threadIdx.x == 0) {
    const float ideal = 0.125f, li = logf(0.125f);
    float s = 0.f;
    for (int e = 0; e < 8; ++e) s += ideal * (li - logf(usage[e] + 1e-10f));
    out_aux[0] = s / 8.0f;
  }
}

// ---------------------------------------------------------------------------
// Main expert kernel: 9 slots (A0 A1 B0 B1 C0 C1 D0 D1 S), fused 2-layer MLPs
// ---------------------------------------------------------------------------
__global__ __launch_bounds__(256) void expert_kernel(
    const float* __restrict__ x, const float* __restrict__ gatew,
    float* __restrict__ out, int Btot,
    const float* __restrict__ AW1, const float* __restrict__ Ab1,
    const float* __restrict__ AW2, const float* __restrict__ Ab2,
    const float* __restrict__ BW1, const float* __restrict__ Bb1,
    const float* __restrict__ Bg,  const float* __restrict__ Bbe,
    const float* __restrict__ BW2, const float* __restrict__ Bb2,
    const float* __restrict__ Weff, const float* __restrict__ beff,
    const float* __restrict__ DW1, const float* __restrict__ Db1,
    const float* __restrict__ Dg,  const float* __restrict__ Dbe,
    const float* __restrict__ DW2, const float* __restrict__ Db2,
    const float* __restrict__ SW1, const float* __restrict__ Sb1,
    const float* __restrict__ Sg,  const float* __restrict__ Sbe,
    const float* __restrict__ SW2, const float* __restrict__ Sb2)
{
  __shared__ float xs[TM * XSS];
  __shared__ float hs[TM * HSS];
  __shared__ float gws[TM * 8];
  __shared__ float mus[TM], ivs[TM];

  const int tid = threadIdx.x;
  const int b0  = blockIdx.x * TM;

  for (int i = tid; i < TM * 256; i += 256) {
    int m = i >> 8, k = i & 255;
    xs[m * XSS + k] = x[(size_t)(b0 + m) * 256 + k];
  }
  for (int i = tid; i < TM * 8; i += 256) gws[i] = gatew[(size_t)b0 * 8 + i];
  __syncthreads();

  const int lane = tid & 31, wave = tid >> 5;
  const int ln = lane & 15, hi = lane >> 4;
  const int n = wave * 16 + ln;             // output column 0..127
  const float RS = rsqrtf(1.0f + 1e-5f);

  v8f priv0 = {}; v8f priv1 = {};

  for (int slot = 0; slot < 9; ++slot) {
    const float *W1, *b1, *W2 = nullptr, *b2 = nullptr, *gv = nullptr, *bev = nullptr;
    int act, eidx, doLN;   // act: 0 relu, 1 tanh-affine, 2 gelu-final, 3 silu
    switch (slot) {
      case 0: case 1: {
        int e = slot;
        W1 = AW1 + (size_t)e * 128 * 256; b1 = Ab1 + e * 128;
        W2 = AW2 + (size_t)e * 128 * 128; b2 = Ab2 + e * 128;
        act = 0; eidx = slot; doLN = 0; break; }
      case 2: case 3: {
        int e = slot - 2;
        W1 = BW1 + (size_t)e * 128 * 256; b1 = Bb1 + e * 128;
        gv = Bg + e * 128; bev = Bbe + e * 128;
        W2 = BW2 + (size_t)e * 128 * 128; b2 = Bb2 + e * 128;
        act = 1; eidx = slot; doLN = 0; break; }
      case 4: case 5: {
        int e = slot - 4;
        W1 = Weff + (size_t)e * 128 * 256; b1 = beff + e * 128;
        act = 2; eidx = slot; doLN = 0; break; }
      case 6: case 7: {
        int e = slot - 6;
        W1 = DW1 + (size_t)e * 128 * 256; b1 = Db1 + e * 128;
        gv = Dg + e * 128; bev = Dbe + e * 128;
        W2 = DW2 + (size_t)e * 128 * 128; b2 = Db2 + e * 128;
        act = 3; eidx = slot; doLN = 1; break; }
      default: {
        W1 = SW1; b1 = Sb1; gv = Sg; bev = Sbe; W2 = SW2; b2 = Sb2;
        act = 0; eidx = -1; doLN = 1; break; }
    }

    // ---- GEMM1: (32x256) x (256x128), fp32 WMMA 16x16x4, K loop of 64 ----
    v8f c0 = {}; v8f c1 = {};
    const float* wr = W1 + (size_t)n * 256;
    #pragma unroll 4
    for (int k0 = 0; k0 < 256; k0 += 4) {
      const int ko = k0 + 2 * hi;
      v2f a0 = *(const v2f*)&xs[ln * XSS + ko];
      v2f a1 = *(const v2f*)&xs[(16 + ln) * XSS + ko];
      v2f bb = *(const v2f*)&wr[ko];
      c0 = wmma4(a0, bb, c0);
      c1 = wmma4(a1, bb, c1);
    }
    const float bias1 = b1[n];

    if (act == 2) {   // C experts: gelu is the final output, no second layer
      for (int r = 0; r < 8; ++r) {
        int m0 = r + 8 * hi;
        float z0 = c0[r] + bias1, z1 = c1[r] + bias1;
        float g0 = 0.5f * z0 * (1.0f + erff(z0 * 0.70710678f));
        float g1 = 0.5f * z1 * (1.0f + erff(z1 * 0.70710678f));
        priv0[r] += gws[m0 * 8 + eidx] * g0;
        priv1[r] += gws[(16 + m0) * 8 + eidx] * g1;
      }
      continue;
    }

    __syncthreads();    // previous slot's GEMM2 readers done before hs overwrite
    {
      const float gn  = (act == 1) ? gv[n]  : 0.f;
      const float ben = (act == 1) ? bev[n] : 0.f;
      for (int r = 0; r < 8; ++r) {
        int m0 = r + 8 * hi;
        float z0 = c0[r] + bias1, z1 = c1[r] + bias1, h0, h1;
        if (act == 0)      { h0 = fmaxf(z0, 0.f); h1 = fmaxf(z1, 0.f); }
        else if (act == 1) { h0 = tanhf(z0) * RS * gn + ben;
                             h1 = tanhf(z1) * RS * gn + ben; }
        else               { h0 = z0 / (1.f + expf(-z0));
                             h1 = z1 / (1.f + expf(-z1)); }
        hs[m0 * HSS + n]        = h0;
        hs[(16 + m0) * HSS + n] = h1;
      }
    }
    __syncthreads();

    if (doLN) {
      if (tid < TM) {
        float s = 0.f, s2 = 0.f;
        for (int o = 0; o < 128; ++o) { float h = hs[tid * HSS + o]; s += h; s2 += h * h; }
        float mu  = s * (1.f / 128.f);
        float var = s2 * (1.f / 128.f) - mu * mu;
        mus[tid] = mu; ivs[tid] = rsqrtf(var + 1e-5f);
      }
      __syncthreads();
      for (int i = tid; i < TM * 128; i += 256) {
        int m = i >> 7, o = i & 127;
        hs[m * HSS + o] = (hs[m * HSS + o] - mus[m]) * ivs[m] * gv[o] + bev[o];
      }
      __syncthreads();
    }

    // ---- GEMM2: (32x128) x (128x128), K loop of 32 ----
    v8f d0 = {}; v8f d1 = {};
    const float* w2r = W2 + (size_t)n * 128;
    #pragma unroll 4
    for (int k0 = 0; k0 < 128; k0 += 4) {
      const int ko = k0 + 2 * hi;
      v2f a0 = *(const v2f*)&hs[ln * HSS + ko];
      v2f a1 = *(const v2f*)&hs[(16 + ln) * HSS + ko];
      v2f bb = *(const v2f*)&w2r[ko];
      d0 = wmma4(a0, bb, d0);
      d1 = wmma4(a1, bb, d1);
    }
    const float bias2 = b2[n];
    if (slot == 8) {          // shared expert -> direct output
      for (int r = 0; r < 8; ++r) {
        int m0 = r + 8 * hi;
        out[(size_t)(b0 + m0) * 128 + n]      = d0[r] + bias2;
        out[(size_t)(b0 + 16 + m0) * 128 + n] = d1[r] + bias2;
      }
    } else {
      for (int r = 0; r < 8; ++r) {
        int m0 = r + 8 * hi;
        priv0[r] += gws[m0 * 8 + eidx]        * (d0[r] + bias2);
        priv1[r] += gws[(16 + m0) * 8 + eidx] * (d1[r] + bias2);
      }
    }
  }

  const size_t pbase = (size_t)Btot * 128;
  for (int r = 0; r < 8; ++r) {
    int m0 = r + 8 * hi;
    out[pbase + (size_t)(b0 + m0) * 128 + n]      = priv0[r];
    out[pbase + (size_t)(b0 + 16 + m0) * 128 + n] = priv1[r];
  }
}

// ---------------------------------------------------------------------------
extern "C" void kernel_launch(void* const* d_in, const int* in_sizes, int n_in,
                              void* d_out, int out_size, void* d_ws, size_t ws_size,
                              hipStream_t stream) {
  const float* x    = (const float*)d_in[0];
  const int*   mod  = (const int*)d_in[1];
  const float* AW1  = (const float*)d_in[2];
  const float* Ab1  = (const float*)d_in[3];
  const float* AW2  = (const float*)d_in[4];
  const float* Ab2  = (const float*)d_in[5];
  const float* BW1  = (const float*)d_in[6];
  const float* Bb1  = (const float*)d_in[7];
  const float* Bg   = (const float*)d_in[8];
  const float* Bbe  = (const float*)d_in[9];
  const float* BW2  = (const float*)d_in[10];
  const float* Bb2  = (const float*)d_in[11];
  const float* Ccw  = (const float*)d_in[12];
  const float* Ccb  = (const float*)d_in[13];
  const float* CW   = (const float*)d_in[14];
  const float* Cb   = (const float*)d_in[15];
  const float* DW1  = (const float*)d_in[16];
  const float* Db1  = (const float*)d_in[17];
  const float* Dg   = (const float*)d_in[18];
  const float* Dbe  = (const float*)d_in[19];
  const float* DW2  = (const float*)d_in[20];
  const float* Db2  = (const float*)d_in[21];
  const float* SW1  = (const float*)d_in[22];
  const float* Sb1  = (const float*)d_in[23];
  const float* Sg   = (const float*)d_in[24];
  const float* Sbe  = (const float*)d_in[25];
  const float* SW2  = (const float*)d_in[26];
  const float* Sb2  = (const float*)d_in[27];
  const float* GW1  = (const float*)d_in[28];
  const float* Gb1  = (const float*)d_in[29];
  const float* GW2  = (const float*)d_in[30];
  const float* Gb2  = (const float*)d_in[31];

  const int Btot = in_sizes[0] / 256;
  float* ws    = (float*)d_ws;
  float* gatew = ws;
  float* probs = gatew + (size_t)Btot * 8;
  float* Weff  = probs + (size_t)Btot * 8;
  float* beff  = Weff + 2 * 128 * 256;
  float* usage = beff + 2 * 128;
  float* out   = (float*)d_out;

  prep_c_kernel<<<2, 128, 0, stream>>>(Ccw, Ccb, CW, Cb, Weff, beff);
  gate_kernel<<<Btot / TM, 256, 0, stream>>>(x, mod, GW1, Gb1, GW2, Gb2, gatew, probs);
  usage_kernel<<<8, 256, 0, stream>>>(probs, usage, Btot);
  aux_kernel<<<1, 32, 0, stream>>>(usage, out + (size_t)2 * Btot * 128);
  expert_kernel<<<Btot / TM, 256, 0, stream>>>(
      x, gatew, out, Btot,
      AW1, Ab1, AW2, Ab2,
      BW1, Bb1, Bg, Bbe, BW2, Bb2,
      Weff, beff,
      DW1, Db1, Dg, Dbe, DW2, Db2,
      SW1, Sb1, Sg, Sbe, SW2, Sb2);
}